// ObscuredBirkhoffManifold_88304527606142
// MI455X (gfx1250) — compile-verified
//
#include <hip/hip_runtime.h>

#define NB 16
#define NN 2048
#define ITERS 50
#define CLAMP_MINV 1e-8f

typedef __attribute__((ext_vector_type(2))) float v2f;
typedef __attribute__((ext_vector_type(8))) float v8f;

// ---------------------------------------------------------------------------
// K = exp(T / temperature), written into d_out (reused as the Gibbs kernel).
// float4-vectorized: 16*2048*2048/4 = 16,777,216 float4 elements.
// ---------------------------------------------------------------------------
__global__ void sink_init_exp(const float* __restrict__ T,
                              float* __restrict__ K,
                              const float* __restrict__ temperature) {
    size_t g = (size_t)blockIdx.x * blockDim.x + threadIdx.x;  // float4 index
    float invt = 1.0f / temperature[0];
    float4 t = ((const float4*)T)[g];
    float4 k;
    k.x = expf(t.x * invt);
    k.y = expf(t.y * invt);
    k.z = expf(t.z * invt);
    k.w = expf(t.w * invt);
    ((float4*)K)[g] = k;
}

// u and v (contiguous in ws): 2 * 16 * 2048 = 65536 floats, set to 1.0
__global__ void sink_init_uv(float* __restrict__ uv) {
    int g = blockIdx.x * blockDim.x + threadIdx.x;
    uv[g] = 1.0f;
}

// ---------------------------------------------------------------------------
// Row step: s_i = (K v)_i per row (one wave per row, float4 coalesced loads,
// wave32 shfl_xor reduction), then u_i = target*u_i / clip(u_i*s_i).
// grid = (256, 16), block = 256 (8 waves -> 8 rows per workgroup)
// ---------------------------------------------------------------------------
__global__ void sink_row_update(const float* __restrict__ K,
                                const float* __restrict__ v,
                                float* __restrict__ u,
                                const float* __restrict__ delta_o) {
    int b    = blockIdx.y;
    int wave = threadIdx.x >> 5;
    int lane = threadIdx.x & 31;
    int row  = blockIdx.x * 8 + wave;

    const float4* Kr4 = (const float4*)(K + ((size_t)b * NN + row) * NN);
    const float4* vb4 = (const float4*)(v + (size_t)b * NN);

    float acc = 0.0f;
#pragma unroll 4
    for (int it = 0; it < 16; ++it) {           // 16 * 32 lanes * 4 = 2048 cols
        int idx = it * 32 + lane;
        float4 kk = Kr4[idx];
        float4 vv = vb4[idx];
        acc += kk.x * vv.x + kk.y * vv.y + kk.z * vv.z + kk.w * vv.w;
    }
    // wave32 reduction
    for (int off = 16; off > 0; off >>= 1)
        acc += __shfl_xor(acc, off, 32);

    if (lane == 0) {
        float target = 1.0f - delta_o[0];
        size_t ui = (size_t)b * NN + row;
        float uo = u[ui];
        float rs = fmaxf(uo * acc, CLAMP_MINV);
        u[ui] = uo * target / rs;
    }
}

// ---------------------------------------------------------------------------
// Column step via WMMA: t_j = (K^T u)_j.
// Each wave owns a 16-column strip over a 512-row chunk.  Per 4-row block:
//   B (4x16 f32) = K-tile with u folded in; A = all-ones 16x4 (layout-proof).
//   D = A*B + C accumulates partial column sums; D row 0 (C vgpr0, lanes 0-15)
//   holds t[j0..j0+15].  4 row-chunk waves reduce via LDS, then update v.
// grid = (64, 16), block = 256 (8 waves = 2 strips x 4 row chunks = 32 cols/WG)
// ---------------------------------------------------------------------------
__global__ void sink_col_update(const float* __restrict__ K,
                                const float* __restrict__ u,
                                float* __restrict__ vvec,
                                const float* __restrict__ delta_o) {
    __shared__ float red[8][16];

    int b      = blockIdx.y;
    int tid    = threadIdx.x;
    int wave   = tid >> 5;
    int lane   = tid & 31;
    int strip  = wave & 1;        // which 16-col strip in this WG
    int rc     = wave >> 1;       // row chunk 0..3 (512 rows each)
    int j0     = blockIdx.x * 32 + strip * 16;
    int lane15 = lane & 15;
    int hi     = lane >> 4;       // lanes 16-31 handle K-rows +2/+3 of each block

    const float* Kb = K + (size_t)b * NN * NN;
    const float* ub = u + (size_t)b * NN;

    v8f c = {0.f, 0.f, 0.f, 0.f, 0.f, 0.f, 0.f, 0.f};
    v2f ones;
    ones.x = 1.0f;
    ones.y = 1.0f;

    int rbeg = rc * 512;
    int rend = rbeg + 512;
    for (int i0 = rbeg; i0 < rend; i0 += 4) {
        // wave-uniform load of u[i0..i0+3] (scalar path)
        float4 uq = *(const float4*)(ub + i0);
        float ua = hi ? uq.z : uq.x;
        float ub2 = hi ? uq.w : uq.y;
        int r0 = i0 + hi * 2;
        // two coalesced 64B-per-half-wave row segments of K, u folded in
        float b0 = Kb[(size_t)r0 * NN + j0 + lane15] * ua;
        float b1 = Kb[(size_t)(r0 + 1) * NN + j0 + lane15] * ub2;
        v2f Bm;
        Bm.x = b0;
        Bm.y = b1;
        // D = ones(16x4) * B(4x16) + C  ==> column sums of B accumulate in C
        c = __builtin_amdgcn_wmma_f32_16x16x4_f32(
                false, ones, false, Bm, (short)0, c, false, false);
    }

    // C vgpr0, lanes 0-15: D[M=0][N=lane] = partial t[j0 + lane]
    if (lane < 16) red[wave][lane] = c[0];
    __syncthreads();

    if (tid < 32) {
        int s = tid >> 4, l = tid & 15;
        float t = red[s][l] + red[s + 2][l] + red[s + 4][l] + red[s + 6][l];
        int j = blockIdx.x * 32 + tid;
        float target = 1.0f - delta_o[0];
        size_t vi = (size_t)b * NN + j;
        float vo = vvec[vi];
        float cs = fmaxf(vo * t, CLAMP_MINV);
        vvec[vi] = vo * target / cs;
    }
}

// ---------------------------------------------------------------------------
// out_ij = u_i * K_ij * v_j   (in place on d_out)
// ---------------------------------------------------------------------------
__global__ void sink_finalize(float* __restrict__ K,
                              const float* __restrict__ u,
                              const float* __restrict__ v) {
    size_t g = (size_t)blockIdx.x * blockDim.x + threadIdx.x;  // float4 index
    size_t rowg = g >> 9;            // 512 float4 per row
    int    c4   = (int)(g & 511);
    size_t b    = rowg >> 11;        // 2048 rows per batch
    int    r    = (int)(rowg & 2047);

    float  uu = u[b * NN + r];
    float4 vv = ((const float4*)(v + b * NN))[c4];
    float4 kk = ((float4*)K)[g];
    kk.x *= uu * vv.x;
    kk.y *= uu * vv.y;
    kk.z *= uu * vv.z;
    kk.w *= uu * vv.w;
    ((float4*)K)[g] = kk;
}

// ---------------------------------------------------------------------------
extern "C" void kernel_launch(void* const* d_in, const int* in_sizes, int n_in,
                              void* d_out, int out_size, void* d_ws, size_t ws_size,
                              hipStream_t stream) {
    const float* T    = (const float*)d_in[0];
    const float* temp = (const float*)d_in[1];
    const float* delt = (const float*)d_in[2];

    float* K = (float*)d_out;          // 16*2048*2048 f32 == out buffer, reused
    float* u = (float*)d_ws;           // [16][2048]
    float* v = u + (size_t)NB * NN;    // [16][2048]   (total ws use: 256 KB)

    // K = exp(T / temperature); u = v = 1
    sink_init_exp<<<65536, 256, 0, stream>>>(T, K, temp);
    sink_init_uv<<<256, 256, 0, stream>>>(u);   // covers u and v (65536 floats)

    for (int it = 0; it < ITERS; ++it) {
        sink_row_update<<<dim3(256, NB), 256, 0, stream>>>(K, v, u, delt);
        sink_col_update<<<dim3(64, NB), 256, 0, stream>>>(K, u, v, delt);
    }

    sink_finalize<<<65536, 256, 0, stream>>>(K, u, v);
}